// DMMPyTorchBaseModel_66108136620479
// MI455X (gfx1250) — compile-verified
//
#include <hip/hip_runtime.h>

typedef float v2f __attribute__((ext_vector_type(2)));
typedef float v8f __attribute__((ext_vector_type(8)));

// D = A(16x4 fp32) x B(4x16 fp32) + C(16x16 fp32)
#define WMMA_F32(a, b, c) \
  __builtin_amdgcn_wmma_f32_16x16x4_f32(false, (a), false, (b), (short)0, (c), false, false)

// Optimizer-level ordering for cross-lane LDS reuse within a single wave.
// LDS is in-order per wave in hardware; this just stops the compiler from
// reordering/hoisting ds ops. Wavefront scope => no heavy cache ops emitted.
#define WAVE_LDS_FENCE() __builtin_amdgcn_fence(__ATOMIC_ACQ_REL, "wavefront")

__device__ __forceinline__ float wave_max32(float v) {
#pragma unroll
  for (int off = 16; off > 0; off >>= 1) v = fmaxf(v, __shfl_xor(v, off, 32));
  return v;
}
__device__ __forceinline__ float wave_sum32(float v) {
#pragma unroll
  for (int off = 16; off > 0; off >>= 1) v += __shfl_xor(v, off, 32);
  return v;
}

// ---------------------------------------------------------------------------
// K0: one wave. A = softmax(T, rows); log_pi = log_softmax(pi);
//     alpha0 = exp(log_pdf[:,0] + log_pi - m0), scalar log-scale s0 = m0.
// ws layout (floats): [0..1023]=A, [1024..1055]=alpha0, [1056]=s0
// ---------------------------------------------------------------------------
__global__ void k0_prep(const float* __restrict__ lp, const float* __restrict__ pi,
                        const float* __restrict__ T, float* __restrict__ w, int N) {
  const int lane = threadIdx.x;  // row i
  float row[32];
  float m = -3.0e38f;
#pragma unroll
  for (int j = 0; j < 32; ++j) { row[j] = T[lane * 32 + j]; m = fmaxf(m, row[j]); }
  float s = 0.0f;
#pragma unroll
  for (int j = 0; j < 32; ++j) { row[j] = __expf(row[j] - m); s += row[j]; }
  const float inv = 1.0f / s;
#pragma unroll
  for (int j = 0; j < 32; ++j) w[lane * 32 + j] = row[j] * inv;

  // log_softmax(pi)
  const float p  = pi[lane];
  const float mp = wave_max32(p);
  const float se = wave_sum32(__expf(p - mp));
  const float lsp = p - mp - __logf(se);

  // alpha0 (normalized linear) + scale
  const float a  = lp[(size_t)lane * N] + lsp;
  const float ma = wave_max32(a);
  w[1024 + lane] = __expf(a - ma);
  if (lane == 0) w[1056] = ma;
}

// ---------------------------------------------------------------------------
// K1: one wave per time-chunk. Propagate 32x32 fp32 matrix B <- diag(p_t)*A*B
//     via V_WMMA_F32_16X16X4_F32, LDS bounce for D->B operand relayout,
//     rescale by 1/max every 8 steps (log accumulated).
// ---------------------------------------------------------------------------
__global__ void __launch_bounds__(128) k1_chunks(const float* __restrict__ lp,
                                                 const float* __restrict__ w,
                                                 float* __restrict__ cM, float* __restrict__ cS,
                                                 int N, int L) {
  __shared__ float smem[4][32][32];
  const int warp = threadIdx.x >> 5, lane = threadIdx.x & 31;
  const int h = lane >> 4, c16 = lane & 15;
  const int c = blockIdx.x * 4 + warp;

  // Preload constant A in A-operand layout: lane holds rows M = c16 (+16*Mt),
  // K = Kt*4 + 2*h + v  (v = vector component).
  v2f a[2][8];
#pragma unroll
  for (int Mt = 0; Mt < 2; ++Mt)
#pragma unroll
    for (int Kt = 0; Kt < 8; ++Kt) {
      const int r = Mt * 16 + c16, col = Kt * 4 + 2 * h;
      v2f t; t[0] = w[r * 32 + col]; t[1] = w[r * 32 + col + 1];
      a[Mt][Kt] = t;
    }

  // Accumulator starts as identity (in LDS, row-major).
#pragma unroll
  for (int r = 0; r < 32; ++r) smem[warp][r][lane] = (r == lane) ? 1.0f : 0.0f;
  WAVE_LDS_FENCE();

  const int t0 = 1 + c * L;
  const int t1 = min(1 + (c + 1) * L, N);
  const float* lprow0 = lp + (size_t)c16 * N;
  const float* lprow1 = lp + (size_t)(16 + c16) * N;
  float logacc = 0.0f;
  int since = 0;

  for (int t = t0; t < t1; ++t) {
    // B operand: tile(Kt,Nt): K = Kt*4 + 2*h + v, Ncol = Nt*16 + c16
    v2f b[8][2];
#pragma unroll
    for (int Kt = 0; Kt < 8; ++Kt)
#pragma unroll
      for (int Nt = 0; Nt < 2; ++Nt) {
        v2f tb;
        tb[0] = smem[warp][Kt * 4 + 2 * h + 0][Nt * 16 + c16];
        tb[1] = smem[warp][Kt * 4 + 2 * h + 1][Nt * 16 + c16];
        b[Kt][Nt] = tb;
      }

    const float p0 = __expf(lprow0[t]);   // emission for rows 0..15 (this lane's row c16)
    const float p1 = __expf(lprow1[t]);   // emission for rows 16..31

    // Fold diag(p_t) into A rows (A-operand is row-per-lane -> scalar mul).
    v2f ap[2][8];
#pragma unroll
    for (int Kt = 0; Kt < 8; ++Kt) { ap[0][Kt] = a[0][Kt] * p0; ap[1][Kt] = a[1][Kt] * p1; }

    v8f d[2][2];
#pragma unroll
    for (int Mt = 0; Mt < 2; ++Mt)
#pragma unroll
      for (int Nt = 0; Nt < 2; ++Nt) {
        v8f acc = {};
#pragma unroll
        for (int Kt = 0; Kt < 8; ++Kt) acc = WMMA_F32(ap[Mt][Kt], b[Kt][Nt], acc);
        d[Mt][Nt] = acc;
      }

    if (++since == 8 || t == t1 - 1) {
      float m = 0.0f;  // entries are non-negative
#pragma unroll
      for (int Mt = 0; Mt < 2; ++Mt)
#pragma unroll
        for (int Nt = 0; Nt < 2; ++Nt)
#pragma unroll
          for (int v = 0; v < 8; ++v) m = fmaxf(m, d[Mt][Nt][v]);
      m = fmaxf(wave_max32(m), 1e-37f);
      const float inv = 1.0f / m;
#pragma unroll
      for (int Mt = 0; Mt < 2; ++Mt)
#pragma unroll
        for (int Nt = 0; Nt < 2; ++Nt) d[Mt][Nt] = d[Mt][Nt] * inv;
      logacc += __logf(m);
      since = 0;
      __builtin_prefetch(lprow0 + t + 64, 0, 0);
      __builtin_prefetch(lprow1 + t + 64, 0, 0);
    }

    // D (row M = Mt*16 + v + 8*h, col = Nt*16 + c16) back to LDS row-major.
#pragma unroll
    for (int Mt = 0; Mt < 2; ++Mt)
#pragma unroll
      for (int Nt = 0; Nt < 2; ++Nt)
#pragma unroll
        for (int v = 0; v < 8; ++v)
          smem[warp][Mt * 16 + v + 8 * h][Nt * 16 + c16] = d[Mt][Nt][v];
    WAVE_LDS_FENCE();
  }

#pragma unroll
  for (int r = 0; r < 32; ++r) cM[(size_t)c * 1024 + r * 32 + lane] = smem[warp][r][lane];
  if (lane == 0) cS[c] = logacc;
}

// ---------------------------------------------------------------------------
// K2: one wave combines 32 consecutive chunk matrices (time order) by WMMA.
// ---------------------------------------------------------------------------
__global__ void __launch_bounds__(128) k2_combine(const float* __restrict__ cM,
                                                  const float* __restrict__ cS,
                                                  float* __restrict__ l2M, float* __restrict__ l2S) {
  __shared__ float smem[4][32][32];
  const int warp = threadIdx.x >> 5, lane = threadIdx.x & 31;
  const int h = lane >> 4, c16 = lane & 15;
  const int g = blockIdx.x * 4 + warp;

#pragma unroll
  for (int r = 0; r < 32; ++r) smem[warp][r][lane] = (r == lane) ? 1.0f : 0.0f;
  WAVE_LDS_FENCE();

  float logacc = 0.0f;
  int since = 0;
  for (int j = g * 32; j < g * 32 + 32; ++j) {
    logacc += cS[j];
    const float* M = cM + (size_t)j * 1024;
    v2f a[2][8], b[8][2];
#pragma unroll
    for (int Mt = 0; Mt < 2; ++Mt)
#pragma unroll
      for (int Kt = 0; Kt < 8; ++Kt) {
        const int r = Mt * 16 + c16, col = Kt * 4 + 2 * h;
        v2f t; t[0] = M[r * 32 + col]; t[1] = M[r * 32 + col + 1];
        a[Mt][Kt] = t;
      }
#pragma unroll
    for (int Kt = 0; Kt < 8; ++Kt)
#pragma unroll
      for (int Nt = 0; Nt < 2; ++Nt) {
        v2f tb;
        tb[0] = smem[warp][Kt * 4 + 2 * h + 0][Nt * 16 + c16];
        tb[1] = smem[warp][Kt * 4 + 2 * h + 1][Nt * 16 + c16];
        b[Kt][Nt] = tb;
      }
    v8f d[2][2];
#pragma unroll
    for (int Mt = 0; Mt < 2; ++Mt)
#pragma unroll
      for (int Nt = 0; Nt < 2; ++Nt) {
        v8f acc = {};
#pragma unroll
        for (int Kt = 0; Kt < 8; ++Kt) acc = WMMA_F32(a[Mt][Kt], b[Kt][Nt], acc);
        d[Mt][Nt] = acc;
      }
    if (++since == 8 || j == g * 32 + 31) {
      float m = 0.0f;
#pragma unroll
      for (int Mt = 0; Mt < 2; ++Mt)
#pragma unroll
        for (int Nt = 0; Nt < 2; ++Nt)
#pragma unroll
          for (int v = 0; v < 8; ++v) m = fmaxf(m, d[Mt][Nt][v]);
      m = fmaxf(wave_max32(m), 1e-37f);
      const float inv = 1.0f / m;
#pragma unroll
      for (int Mt = 0; Mt < 2; ++Mt)
#pragma unroll
        for (int Nt = 0; Nt < 2; ++Nt) d[Mt][Nt] = d[Mt][Nt] * inv;
      logacc += __logf(m);
      since = 0;
    }
#pragma unroll
    for (int Mt = 0; Mt < 2; ++Mt)
#pragma unroll
      for (int Nt = 0; Nt < 2; ++Nt)
#pragma unroll
        for (int v = 0; v < 8; ++v)
          smem[warp][Mt * 16 + v + 8 * h][Nt * 16 + c16] = d[Mt][Nt][v];
    WAVE_LDS_FENCE();
  }
#pragma unroll
  for (int r = 0; r < 32; ++r) l2M[(size_t)g * 1024 + r * 32 + lane] = smem[warp][r][lane];
  if (lane == 0) l2S[g] = logacc;
}

// ---------------------------------------------------------------------------
// K3: one wave. alpha through G2 level-2 matrices (matvec), final logsumexp.
// ---------------------------------------------------------------------------
__global__ void k3_final(const float* __restrict__ w, const float* __restrict__ l2M,
                         const float* __restrict__ l2S, float* __restrict__ out, int G2) {
  __shared__ float sa[32];
  const int lane = threadIdx.x;
  float alpha = w[1024 + lane];
  float total = w[1056];
  for (int g = 0; g < G2; ++g) {
    total += l2S[g];
    sa[lane] = alpha;  // single wave: DS in-order; fences stop compiler reordering
    WAVE_LDS_FENCE();
    const float* Mrow = l2M + (size_t)g * 1024 + lane * 32;
    float acc = 0.0f;
#pragma unroll
    for (int j = 0; j < 32; ++j) acc += Mrow[j] * sa[j];
    WAVE_LDS_FENCE();
    float m = fmaxf(wave_max32(acc), 1e-37f);
    alpha = acc / m;
    total += __logf(m);
  }
  const float s = wave_sum32(alpha);
  if (lane == 0) out[0] = __logf(s) + total;
}

// ---------------------------------------------------------------------------
static inline long long ws_need_floats(long long C) {
  const long long g2 = C / 32;
  return 1088 + C * 1024 + C + g2 * 1024 + g2;
}

extern "C" void kernel_launch(void* const* d_in, const int* in_sizes, int n_in,
                              void* d_out, int out_size, void* d_ws, size_t ws_size,
                              hipStream_t stream) {
  (void)n_in; (void)out_size;
  const float* log_pdf = (const float*)d_in[0];
  const float* pi      = (const float*)d_in[1];
  const float* T       = (const float*)d_in[2];
  float* w   = (float*)d_ws;
  float* out = (float*)d_out;

  const int N = in_sizes[0] / 32;  // log_pdf is [32, N]

  int C = 4096;  // chunks (multiple of 128)
  while (C > 128 && (size_t)(ws_need_floats(C) * 4LL) > ws_size) C >>= 1;
  const int G2 = C / 32;
  const int L  = (N - 1 + C - 1) / C;  // timesteps per chunk

  float* chunkM = w + 1088;
  float* chunkS = chunkM + (size_t)C * 1024;
  float* l2M    = chunkS + C;
  float* l2S    = l2M + (size_t)G2 * 1024;

  k0_prep<<<1, 32, 0, stream>>>(log_pdf, pi, T, w, N);
  k1_chunks<<<C / 4, 128, 0, stream>>>(log_pdf, w, chunkM, chunkS, N, L);
  k2_combine<<<G2 / 4, 128, 0, stream>>>(chunkM, chunkS, l2M, l2S);
  k3_final<<<1, 32, 0, stream>>>(w, l2M, l2S, out, G2);
}